// GraphRfi_7997229105852
// MI455X (gfx1250) — compile-verified
//
#include <hip/hip_runtime.h>

typedef __attribute__((ext_vector_type(2))) float v2f;
typedef __attribute__((ext_vector_type(8))) float v8f;

#define THREADS 256

// ---------------------------------------------------------------------------
// 1) deg[n] = 1 (self loop), out = 0   (must re-zero d_out each call)
// ---------------------------------------------------------------------------
__global__ __launch_bounds__(THREADS) void k_init(float* __restrict__ deg,
                                                  float4* __restrict__ out4,
                                                  int n_nodes, int n_out4) {
    int i = blockIdx.x * THREADS + threadIdx.x;
    if (i < n_nodes) deg[i] = 1.0f;
    if (i < n_out4)  out4[i] = make_float4(0.f, 0.f, 0.f, 0.f);
}

// ---------------------------------------------------------------------------
// 2) degree at dst
// ---------------------------------------------------------------------------
__global__ __launch_bounds__(THREADS) void k_deg(const int* __restrict__ dstE,
                                                 float* __restrict__ deg,
                                                 int n_edges) {
    int e = blockIdx.x * THREADS + threadIdx.x;
    if (e < n_edges) atomicAdd(&deg[dstE[e]], 1.0f);
}

// ---------------------------------------------------------------------------
// 3) dinv = deg > 0 ? rsqrt(deg) : 0
// ---------------------------------------------------------------------------
__global__ __launch_bounds__(THREADS) void k_dinv(const float* __restrict__ deg,
                                                  float* __restrict__ dinv, int n) {
    int i = blockIdx.x * THREADS + threadIdx.x;
    if (i < n) {
        float d = deg[i];
        dinv[i] = (d > 0.f) ? rsqrtf(d) : 0.f;
    }
}

// ---------------------------------------------------------------------------
// 4) h = x @ W via V_WMMA_F32_16X16X4_F32.
//    One wave -> 16x64 output tile, K=64 in 16 steps of 4, 4 column tiles.
//    A layout (16x4 f32): lane(l): M = l%16; half=l/16 -> K base +2;
//                         vgpr v -> K = 4*ks + 2*half + v.
//    B layout (4x16 f32): N = l%16; K = 4*ks + 2*half + v  (W^T in LDS).
//    C layout (16x16 f32): vgpr r -> M = r + 8*half, N = l%16.
// ---------------------------------------------------------------------------
__global__ __launch_bounds__(THREADS) void k_gemm(const float* __restrict__ x,
                                                  const float* __restrict__ W,
                                                  float* __restrict__ h,
                                                  int n_nodes) {
    __shared__ float wt[64 * 64];              // wt[n*64 + k] = W[k*64 + n]
    int tid = threadIdx.x;
    for (int i = tid; i < 64 * 64; i += THREADS)
        wt[i] = W[((i & 63) << 6) + (i >> 6)];
    __syncthreads();

    int wave = tid >> 5;
    int lane = tid & 31;
    int half = lane >> 4;                      // 0 or 1
    int m    = lane & 15;
    int row_base = ((blockIdx.x << 3) + wave) << 4;
    if (row_base >= n_nodes) return;           // uniform per wave -> EXEC all-1s

    const float* xr = x + (size_t)(row_base + m) * 64;
    const float* w0 = wt + (size_t)(0  + m) * 64;
    const float* w1 = wt + (size_t)(16 + m) * 64;
    const float* w2 = wt + (size_t)(32 + m) * 64;
    const float* w3 = wt + (size_t)(48 + m) * 64;

    v8f c0 = {}, c1 = {}, c2 = {}, c3 = {};
#pragma unroll
    for (int ks = 0; ks < 16; ++ks) {
        int kb = ks * 4 + half * 2;
        v2f a  = {xr[kb], xr[kb + 1]};
        v2f b0 = {w0[kb], w0[kb + 1]};
        v2f b1 = {w1[kb], w1[kb + 1]};
        v2f b2 = {w2[kb], w2[kb + 1]};
        v2f b3 = {w3[kb], w3[kb + 1]};
        c0 = __builtin_amdgcn_wmma_f32_16x16x4_f32(false, a, false, b0, (short)0, c0, false, false);
        c1 = __builtin_amdgcn_wmma_f32_16x16x4_f32(false, a, false, b1, (short)0, c1, false, false);
        c2 = __builtin_amdgcn_wmma_f32_16x16x4_f32(false, a, false, b2, (short)0, c2, false, false);
        c3 = __builtin_amdgcn_wmma_f32_16x16x4_f32(false, a, false, b3, (short)0, c3, false, false);
    }

#pragma unroll
    for (int r = 0; r < 8; ++r) {
        size_t row = (size_t)row_base + (size_t)(half * 8 + r);
        float* hr = h + row * 64 + m;
        hr[0]  = c0[r];
        hr[16] = c1[r];
        hr[32] = c2[r];
        hr[48] = c3[r];
    }
}

// ---------------------------------------------------------------------------
// 5) out[dst] += h[src] * dinv[src]*dinv[dst]   (16 lanes per edge, 4 ch each)
// ---------------------------------------------------------------------------
__global__ __launch_bounds__(THREADS) void k_scatter(const int* __restrict__ srcE,
                                                     const int* __restrict__ dstE,
                                                     const float* __restrict__ dinv,
                                                     const float* __restrict__ h,
                                                     float* __restrict__ out,
                                                     int n_edges) {
    int gid = blockIdx.x * THREADS + threadIdx.x;
    int e  = gid >> 4;
    int c4 = (gid & 15) << 2;
    if (e >= n_edges) return;
    int s = srcE[e];
    int d = dstE[e];
    float nrm = dinv[s] * dinv[d];
    const float4 hv = *(const float4*)(h + (size_t)s * 64 + c4);
    float* op = out + (size_t)d * 64 + c4;
    atomicAdd(op + 0, hv.x * nrm);
    atomicAdd(op + 1, hv.y * nrm);
    atomicAdd(op + 2, hv.z * nrm);
    atomicAdd(op + 3, hv.w * nrm);
}

// ---------------------------------------------------------------------------
// 6) out = relu(out + h[n]*dinv[n]^2 + b)   (self-loop fused here)
// ---------------------------------------------------------------------------
__global__ __launch_bounds__(THREADS) void k_final(const float* __restrict__ h,
                                                   const float* __restrict__ dinv,
                                                   const float* __restrict__ bias,
                                                   float* __restrict__ out,
                                                   int n_nodes) {
    int gid = blockIdx.x * THREADS + threadIdx.x;
    int n  = gid >> 4;
    int c4 = (gid & 15) << 2;
    if (n >= n_nodes) return;
    float di = dinv[n];
    float sl = di * di;
    const float4 hv = *(const float4*)(h + (size_t)n * 64 + c4);
    const float4 bv = *(const float4*)(bias + c4);
    float4 o = *(float4*)(out + (size_t)n * 64 + c4);
    o.x = fmaxf(o.x + hv.x * sl + bv.x, 0.f);
    o.y = fmaxf(o.y + hv.y * sl + bv.y, 0.f);
    o.z = fmaxf(o.z + hv.z * sl + bv.z, 0.f);
    o.w = fmaxf(o.w + hv.w * sl + bv.w, 0.f);
    *(float4*)(out + (size_t)n * 64 + c4) = o;
}

// ---------------------------------------------------------------------------
extern "C" void kernel_launch(void* const* d_in, const int* in_sizes, int n_in,
                              void* d_out, int out_size, void* d_ws, size_t ws_size,
                              hipStream_t stream) {
    const float* x    = (const float*)d_in[0];   // [N, 64]
    const int*   edge = (const int*)d_in[1];     // [2, E] row-major
    const float* W    = (const float*)d_in[2];   // [64, 64]
    const float* bias = (const float*)d_in[3];   // [64]

    const int n_nodes = in_sizes[0] / 64;
    const int n_edges = in_sizes[1] / 2;
    const int* srcE = edge;
    const int* dstE = edge + n_edges;

    float* out = (float*)d_out;

    // Workspace layout: h [N*64] | deg [N] | dinv [N]
    char* ws = (char*)d_ws;
    float* h    = (float*)ws;
    float* deg  = (float*)(ws + (size_t)n_nodes * 64 * sizeof(float));
    float* dinv = deg + n_nodes;

    const int n_out4 = n_nodes * 16;   // N*64/4 float4s

    // 1) init deg=1, out=0
    {
        int total = n_out4;  // n_out4 >= n_nodes
        int grid = (total + THREADS - 1) / THREADS;
        k_init<<<grid, THREADS, 0, stream>>>(deg, (float4*)out, n_nodes, n_out4);
    }
    // 2) degree
    {
        int grid = (n_edges + THREADS - 1) / THREADS;
        k_deg<<<grid, THREADS, 0, stream>>>(dstE, deg, n_edges);
    }
    // 3) dinv
    {
        int grid = (n_nodes + THREADS - 1) / THREADS;
        k_dinv<<<grid, THREADS, 0, stream>>>(deg, dinv, n_nodes);
    }
    // 4) h = x @ W  (WMMA)
    {
        int tiles = (n_nodes + 15) / 16;
        int grid = (tiles + 7) / 8;          // 8 waves / block, 1 tile / wave
        k_gemm<<<grid, THREADS, 0, stream>>>(x, W, h, n_nodes);
    }
    // 5) edge scatter
    {
        long long total = (long long)n_edges * 16;
        int grid = (int)((total + THREADS - 1) / THREADS);
        k_scatter<<<grid, THREADS, 0, stream>>>(srcE, dstE, dinv, h, out, n_edges);
    }
    // 6) finalize (self-loop + bias + relu)
    {
        int total = n_nodes * 16;
        int grid = (total + THREADS - 1) / THREADS;
        k_final<<<grid, THREADS, 0, stream>>>(h, dinv, bias, out, n_nodes);
    }
}